// EnhancedReconstructionLoss_86397562126740
// MI455X (gfx1250) — compile-verified
//
#include <hip/hip_runtime.h>
#include <stdint.h>

// ---------------------------------------------------------------------------
// Fused MSE + SSIM loss for (32,3,512,512) fp32 tensors.
// Memory-bound: 201 MB compulsory reads @ 23.3 TB/s ~= 8.6 us floor.
// Single pass over HBM; tiles staged in LDS with CDNA5 async global->LDS
// b128 copies (ASYNCcnt), separable 3x3 sums, hierarchical f64 reduction.
// ---------------------------------------------------------------------------

#define IMG_W   512
#define IMG_H   512
#define TILE_W  128         // output tile width  per block
#define TILE_H  16          // output tile height per block
#define LDS_W   148         // LDS row stride in floats (592 B, 16B-aligned)
#define LDS_ROWS 18         // TILE_H + 2 halo rows
#define NGRP    36          // b128 groups per row: cols [tx0-8, tx0+136)
#define NTHREADS 256        // 8 waves (wave32)

#define MSE_W   0.8
#define SSIM_W  0.2
#define C1_     1.0e-4f     // 0.01^2
#define C2_     9.0e-4f     // 0.03^2
#define EPS_    1.0e-8f

// CDNA5 async copy: global memory -> LDS, 16 bytes, tracked on ASYNCcnt.
// Low 32 bits of a flat LDS pointer are the DS byte offset (ISA 10.2).
__device__ __forceinline__ void async_ld_b128(void* lds_dst, const void* gsrc) {
    uint32_t loff = (uint32_t)(uintptr_t)lds_dst;
    unsigned long long ga = (unsigned long long)(uintptr_t)gsrc;
    asm volatile("global_load_async_to_lds_b128 %0, %1, off"
                 :: "v"(loff), "v"(ga)
                 : "memory");
}

__device__ __forceinline__ void wait_asynccnt0() {
    asm volatile("s_wait_asynccnt 0" ::: "memory");
}

__global__ __launch_bounds__(NTHREADS)
void fused_mse_ssim_kernel(const float* __restrict__ x,
                           const float* __restrict__ y,
                           double* __restrict__ acc) {
    __shared__ __align__(16) float sx[LDS_ROWS * LDS_W];
    __shared__ __align__(16) float sy[LDS_ROWS * LDS_W];
    __shared__ float wred[16];  // [0..7] ssim partials, [8..15] mse partials

    const int tid   = threadIdx.x;
    const int tx0   = blockIdx.x * TILE_W;   // output tile origin (col)
    const int ty0   = blockIdx.y * TILE_H;   // output tile origin (row)
    const int plane = blockIdx.z;

    const float* xp = x + (size_t)plane * (IMG_W * IMG_H);
    const float* yp = y + (size_t)plane * (IMG_W * IMG_H);

    // Zero-pad semantics (count_include_pad): border tiles pre-zero LDS so
    // skipped (out-of-image) async loads leave zeros behind.  Condition is
    // block-uniform, so the extra barrier is uniform too.
    const bool border = (tx0 == 0) | (ty0 == 0) |
                        (tx0 + TILE_W == IMG_W) | (ty0 + TILE_H == IMG_H);
    if (border) {
        for (int i = tid; i < LDS_ROWS * LDS_W; i += NTHREADS) {
            sx[i] = 0.0f;
            sy[i] = 0.0f;
        }
        __syncthreads();   // ds stores land before async writes
    }

    // Async-load an 18-row x 144-col window: global cols [tx0-8, tx0+136),
    // rows [ty0-1, ty0+17).  Every b128 group is 16B-aligned and is either
    // fully inside or fully outside the image.  Compute reads only lds cols
    // 7..136 (global cols tx0-1 .. tx0+128), never the alignment slack.
    for (int i = tid; i < LDS_ROWS * NGRP; i += NTHREADS) {
        const int r  = i / NGRP;
        const int g  = i % NGRP;
        const int gr = ty0 - 1 + r;
        const int gc = tx0 - 8 + g * 4;
        if ((unsigned)gr < (unsigned)IMG_H && (unsigned)gc < (unsigned)IMG_W) {
            const size_t off = (size_t)gr * IMG_W + gc;
            async_ld_b128(&sx[r * LDS_W + g * 4], xp + off);
            async_ld_b128(&sy[r * LDS_W + g * 4], yp + off);
        }
    }
    wait_asynccnt0();      // each wave waits for its own async copies
    __syncthreads();       // tile now visible to the whole block

    // Each thread: 8 consecutive outputs in one row (16 threads x 16 rows).
    const int orow  = tid >> 4;         // 0..15
    const int ocol0 = (tid & 15) * 8;   // 0,8,...,120

    // Separable 3x3: per column, vertical 3-row sums of x, y, x^2, y^2, xy.
    // 10 columns (8 outputs + 2 halo) amortized across the 8 outputs.
    float csx[10], csy[10], csxx[10], csyy[10], csxy[10];
    float mse_acc = 0.0f;
    #pragma unroll
    for (int c = 0; c < 10; ++c) {
        const int li = orow * LDS_W + ocol0 + 7 + c;   // lds rows orow..orow+2
        const float x0 = sx[li];
        const float x1 = sx[li + LDS_W];
        const float x2 = sx[li + 2 * LDS_W];
        const float y0 = sy[li];
        const float y1 = sy[li + LDS_W];
        const float y2 = sy[li + 2 * LDS_W];
        csx[c]  = x0 + x1 + x2;
        csy[c]  = y0 + y1 + y2;
        csxx[c] = x0 * x0 + x1 * x1 + x2 * x2;
        csyy[c] = y0 * y0 + y1 * y1 + y2 * y2;
        csxy[c] = x0 * y0 + x1 * y1 + x2 * y2;
        if (c >= 1 && c <= 8) {          // center pixel of this column
            const float d = x1 - y1;
            mse_acc += d * d;
        }
    }

    // Horizontal 3-column sums + SSIM rational map.
    const float i9  = 1.0f / 9.0f;
    const float i81 = 1.0f / 81.0f;
    float ssim_acc = 0.0f;
    #pragma unroll
    for (int k = 0; k < 8; ++k) {
        const float Sx  = csx[k]  + csx[k + 1]  + csx[k + 2];
        const float Sy  = csy[k]  + csy[k + 1]  + csy[k + 2];
        const float Sxx = csxx[k] + csxx[k + 1] + csxx[k + 2];
        const float Syy = csyy[k] + csyy[k + 1] + csyy[k + 2];
        const float Sxy = csxy[k] + csxy[k + 1] + csxy[k + 2];

        const float mu_xy = Sx * Sy * i81;                  // mu_x * mu_y
        const float musq  = (Sx * Sx + Sy * Sy) * i81;      // mu_x^2 + mu_y^2
        const float sig_xy = Sxy * i9 - mu_xy;              // sigma_xy
        const float sigsum = (Sxx + Syy) * i9 - musq;       // sig_x^2+sig_y^2

        const float num = (2.0f * mu_xy + C1_) * (2.0f * sig_xy + C2_);
        const float den = (musq + C1_) * (sigsum + C2_) + EPS_;
        ssim_acc += num / den;
    }

    // Hierarchical reduction: wave32 shuffle -> LDS -> f64 global atomics.
    float vs = ssim_acc, vm = mse_acc;
    #pragma unroll
    for (int m = 16; m >= 1; m >>= 1) {
        vs += __shfl_xor(vs, m, 32);
        vm += __shfl_xor(vm, m, 32);
    }
    const int wave = tid >> 5;
    if ((tid & 31) == 0) {
        wred[wave]     = vs;
        wred[8 + wave] = vm;
    }
    __syncthreads();
    if (tid == 0) {
        float bs = 0.0f, bm = 0.0f;
        #pragma unroll
        for (int w = 0; w < 8; ++w) { bs += wred[w]; bm += wred[8 + w]; }
        atomicAdd(&acc[0], (double)bs);   // global_atomic_add_f64
        atomicAdd(&acc[1], (double)bm);
    }
}

__global__ void finalize_kernel(const double* __restrict__ acc,
                                float* __restrict__ out, double inv_n) {
    if (threadIdx.x == 0 && blockIdx.x == 0) {
        const double ssim_mean = acc[0] * inv_n;
        const double mse_mean  = acc[1] * inv_n;
        out[0] = (float)(MSE_W * mse_mean + SSIM_W * (1.0 - ssim_mean));
    }
}

extern "C" void kernel_launch(void* const* d_in, const int* in_sizes, int n_in,
                              void* d_out, int out_size, void* d_ws, size_t ws_size,
                              hipStream_t stream) {
    (void)n_in; (void)out_size; (void)ws_size;
    const float* x = (const float*)d_in[0];
    const float* y = (const float*)d_in[1];
    float* out     = (float*)d_out;
    double* acc    = (double*)d_ws;

    // ws is poisoned with 0xAA before timing; zero our two accumulators.
    hipMemsetAsync(acc, 0, 2 * sizeof(double), stream);

    const int total  = in_sizes[0];                 // 32*3*512*512
    const int planes = total / (IMG_W * IMG_H);     // 96

    dim3 grid(IMG_W / TILE_W, IMG_H / TILE_H, planes);   // 4 x 32 x 96
    fused_mse_ssim_kernel<<<grid, NTHREADS, 0, stream>>>(x, y, acc);

    const double inv_n = 1.0 / (double)total;
    finalize_kernel<<<1, 1, 0, stream>>>(acc, out, inv_n);
}